// CrossTokenMLPAggregator_27092653703661
// MI455X (gfx1250) — compile-verified
//
#include <hip/hip_runtime.h>

typedef __attribute__((ext_vector_type(2))) float v2f;
typedef __attribute__((ext_vector_type(8))) float v8f;

#define BB   2
#define TT   2048
#define HHD  16      // heads
#define HD   1024    // hidden
#define KTOP 8
#define EPSF 1e-8f

// One wave32 per (b, t) query row.
//  Phase 1: head-mean via V_WMMA_F32_16X16X4_F32 (A = const 1/16) -> LDS row
//  Phase 2: top-8 via shuffle-argmax rounds over the LDS row
//  Phase 3: out[row,:] = sum_k w_k * mlp[b, idx_k, :]
__global__ __launch_bounds__(32)
void cross_token_mlp_agg_kernel(const float* __restrict__ mlp,
                                const float* __restrict__ attn,
                                float* __restrict__ out) {
  __shared__ float sums[TT];           // 8 KB: head-mean for this query row

  const int lane = threadIdx.x;        // wave32
  const int row  = blockIdx.x;         // 0 .. B*T-1
  const int b    = row / TT;
  const int t    = row % TT;

  const size_t headStride = (size_t)TT * TT;
  // attn[((b*Hh + h)*T + t)*T + s]
  const float* base = attn + (size_t)b * HHD * headStride + (size_t)t * TT;

  const int p = lane >> 4;             // lane half selects which heads we feed
  const int n = lane & 15;             // B-matrix column (N = lane&15)

  // A(16x4) = 1/16 everywhere -> D[m,n] = (1/16) * sum_k B[k,n] for every m.
  // Symmetric in K, so the exact K-slot ordering of the B layout is irrelevant.
  v2f amat = {0.0625f, 0.0625f};

  // -------- Phase 1: head-mean into LDS (WMMA reduction over 16 heads) -----
  for (int ct = 0; ct < TT / 16; ++ct) {
    const int col = ct * 16 + n;
    v8f acc = {};
#pragma unroll
    for (int hg = 0; hg < 4; ++hg) {   // 4 head-groups of 4 heads (K=4 each)
      v2f bmat;
      bmat[0] = base[(size_t)(4 * hg + p)     * headStride + col];
      bmat[1] = base[(size_t)(4 * hg + 2 + p) * headStride + col];
      acc = __builtin_amdgcn_wmma_f32_16x16x4_f32(
          /*neg_a=*/false, amat, /*neg_b=*/false, bmat,
          /*c_mod=*/(short)0, acc, /*reuse_a=*/false, /*reuse_b=*/false);
    }
    if (lane < 16) sums[col] = acc[0]; // all D rows identical; row 0, N=lane
  }
  __syncthreads();
  asm volatile("s_wait_dscnt 0" ::: "memory");

  // -------- Phase 2: top-8 (values >= 0, mask with -1) ---------------------
  float topv[KTOP];
  int   topi[KTOP];
#pragma unroll 1
  for (int kk = 0; kk < KTOP; ++kk) {
    float bv = -2.0f;
    int   bi = 1 << 30;
    for (int j = 0; j < TT / 32; ++j) {
      const int   c = j * 32 + lane;   // bank-conflict-free stride
      const float v = sums[c];
      if (v > bv || (v == bv && c < bi)) { bv = v; bi = c; }
    }
#pragma unroll
    for (int off = 16; off > 0; off >>= 1) {   // butterfly argmax, all lanes
      const float ov = __shfl_xor(bv, off, 32);
      const int   oi = __shfl_xor(bi, off, 32);
      if (ov > bv || (ov == bv && oi < bi)) { bv = ov; bi = oi; }
    }
    topv[kk] = bv;
    topi[kk] = bi;
    if (lane == 0) sums[bi] = -1.0f;   // mask winner
    asm volatile("s_wait_dscnt 0" ::: "memory");
  }

  float s = 0.0f;
#pragma unroll
  for (int kk = 0; kk < KTOP; ++kk) s += topv[kk];
  s = fmaxf(s, EPSF);
  const float inv = 1.0f / s;
  float w[KTOP];
#pragma unroll
  for (int kk = 0; kk < KTOP; ++kk) w[kk] = topv[kk] * inv;

  // -------- Phase 3: sparse weighted sum over mlp rows (L2-resident) -------
  const float* mlpB   = mlp + (size_t)b * TT * HD;
  float*       outRow = out + (size_t)row * HD;
  for (int it = 0; it < HD / 128; ++it) {      // 32 lanes x float4 = 128/iter
    const int h = it * 128 + lane * 4;
    float4 a4 = make_float4(0.f, 0.f, 0.f, 0.f);
#pragma unroll
    for (int kk = 0; kk < KTOP; ++kk) {
      const float4 m4 = *(const float4*)(mlpB + (size_t)topi[kk] * HD + h);
      a4.x = fmaf(w[kk], m4.x, a4.x);
      a4.y = fmaf(w[kk], m4.y, a4.y);
      a4.z = fmaf(w[kk], m4.z, a4.z);
      a4.w = fmaf(w[kk], m4.w, a4.w);
    }
    *(float4*)(outRow + h) = a4;
  }
}

extern "C" void kernel_launch(void* const* d_in, const int* in_sizes, int n_in,
                              void* d_out, int out_size, void* d_ws, size_t ws_size,
                              hipStream_t stream) {
  (void)in_sizes; (void)n_in; (void)out_size; (void)d_ws; (void)ws_size;
  const float* mlp  = (const float*)d_in[0];   // [2, 2048, 1024] f32
  const float* attn = (const float*)d_in[1];   // [2, 16, 2048, 2048] f32
  float*       out  = (float*)d_out;           // [2, 2048, 1024] f32

  dim3 grid(BB * TT);   // 4096 rows, one wave32 each
  dim3 block(32);
  cross_token_mlp_agg_kernel<<<grid, block, 0, stream>>>(mlp, attn, out);
}